// LaplacianSmoothLoss_30545807409222
// MI455X (gfx1250) — compile-verified
//
#include <hip/hip_runtime.h>
#include <hip/hip_bf16.h>

typedef __attribute__((ext_vector_type(2))) float v2f;
typedef __attribute__((ext_vector_type(8))) float v8f;
typedef __attribute__((ext_vector_type(4))) int   v4i;

#define AS1 __attribute__((address_space(1)))
#define AS3 __attribute__((address_space(3)))

#define EMPTY_SLOT 0xFFFFFFFFFFFFFFFFull

// ---------------------------------------------------------------------------
// Workspace layout (all inside d_ws):
//   [0]                : nbr_sum  float[3V]
//   [3V]               : deg      float[V]
//   [4V]               : acc      float[1]   (scalar accumulator)
//   [256B aligned]     : htab     u64[H]     (open-addressing directed-edge set)
// ---------------------------------------------------------------------------

__device__ __forceinline__ void async_copy16_to_lds(const void* gsrc, void* ldst) {
#if __has_builtin(__builtin_amdgcn_global_load_async_to_lds_b128)
  __builtin_amdgcn_global_load_async_to_lds_b128(
      (AS1 v4i*)(gsrc), (AS3 v4i*)(ldst), 0, 0);
#else
  unsigned long long ga = (unsigned long long)gsrc;
  unsigned int       la = (unsigned int)(unsigned long long)ldst; // low 32b = LDS offset
  asm volatile("global_load_async_to_lds_b128 %0, %1, off"
               :: "v"(la), "v"(ga) : "memory");
#endif
}

__device__ __forceinline__ void wait_asynccnt0() {
#if __has_builtin(__builtin_amdgcn_s_wait_asynccnt)
  __builtin_amdgcn_s_wait_asynccnt(0);
#else
  asm volatile("s_wait_asynccnt 0x0" ::: "memory");
#endif
}

__global__ void lap_init(float* fbase, unsigned long long nfloat,
                         unsigned long long* htab, unsigned long long nslots) {
  unsigned long long idx    = (unsigned long long)blockIdx.x * blockDim.x + threadIdx.x;
  unsigned long long stride = (unsigned long long)gridDim.x * blockDim.x;
  for (unsigned long long i = idx; i < nfloat; i += stride) fbase[i] = 0.0f;
  for (unsigned long long i = idx; i < nslots; i += stride) htab[i] = EMPTY_SLOT;
}

// One thread per face. Full 256-face tiles are staged into LDS with
// GLOBAL_LOAD_ASYNC_TO_LDS_B128 (192 lanes x 16B = 3072B, ASYNCcnt-tracked),
// then each thread emits the 6 directed edges of the reference
// (ii=[b,c,a], jj=[c,a,b], both directions), deduplicates globally via the
// hash set, and scatter-adds deg / neighbor-sum on first insertion only.
// verts + nbr + deg are L2-resident (28 MB total on 192 MB L2).
__global__ void lap_scatter(const float* __restrict__ verts,
                            const int* __restrict__ faces,
                            int F, int Vn,
                            float* __restrict__ nbr, float* __restrict__ deg,
                            unsigned long long* __restrict__ htab,
                            unsigned int hmask) {
  __shared__ __align__(16) int sfaces[768];   // 256 faces * 3 ints
  const int tid = threadIdx.x;
  const int f0  = blockIdx.x * 256;

  int a, b, c;
  if (f0 + 256 <= F) {                        // uniform per block
    if (tid < 192) {
      const char* src = (const char*)faces + (size_t)f0 * 12 + (size_t)tid * 16;
      char*       dst = (char*)sfaces + tid * 16;
      async_copy16_to_lds(src, dst);
    }
    wait_asynccnt0();                         // per-wave async completion
    __syncthreads();                          // cross-wave LDS visibility
    a = sfaces[3 * tid + 0];
    b = sfaces[3 * tid + 1];
    c = sfaces[3 * tid + 2];
  } else {                                    // single partial tail block
    int f = f0 + tid;
    if (f >= F) return;
    a = faces[3 * f + 0];
    b = faces[3 * f + 1];
    c = faces[3 * f + 2];
  }

  int ss[6] = { b, c, a, c, a, b };
  int dd[6] = { c, a, b, b, c, a };
#pragma unroll
  for (int e = 0; e < 6; ++e) {
    unsigned int s = (unsigned int)ss[e];
    unsigned int d = (unsigned int)dd[e];
    unsigned long long key = (unsigned long long)s * (unsigned long long)Vn + d;
    // MurmurHash3 fmix64 finalizer
    unsigned long long h = key;
    h ^= h >> 33; h *= 0xff51afd7ed558ccdull;
    h ^= h >> 33; h *= 0xc4ceb9fe1a85ec53ull;
    h ^= h >> 33;
    unsigned int slot = (unsigned int)h & hmask;
    bool inserted = false;
    for (;;) {
      unsigned long long old = atomicCAS(&htab[slot], EMPTY_SLOT, key);
      if (old == EMPTY_SLOT) { inserted = true; break; } // first occurrence
      if (old == key) break;                             // duplicate edge
      slot = (slot + 1) & hmask;                         // probe
    }
    if (inserted) {
      atomicAdd(&deg[s], 1.0f);
      const float* vd = verts + 3ull * d;
      float*       ns = nbr   + 3ull * s;
      atomicAdd(&ns[0], vd[0]);
      atomicAdd(&ns[1], vd[1]);
      atomicAdd(&ns[2], vd[2]);
    }
  }
}

// Per-vertex Lv = deg*v - nbr_sum, norm, then wave32 reduction:
// v_wmma_f32_16x16x4_f32 with A=ones(16x4) folds the 32 lane values into 16
// column sums (D[m][n] = nrm_n + nrm_{16+n}, replicated over rows), finished
// with 4 xor-shuffles. f32 MACs -> precision identical to scalar adds.
// Blocks are always full (tail lanes contribute 0) so EXEC is all-1s at WMMA.
__global__ void lap_finalize(const float* __restrict__ verts,
                             const float* __restrict__ nbr,
                             const float* __restrict__ deg,
                             float* __restrict__ acc, int Vn) {
  int i = blockIdx.x * blockDim.x + threadIdx.x;
  float nrm = 0.0f;
  if (i < Vn) {
    float dg = deg[i];
    float x = dg * verts[3 * i + 0] - nbr[3 * i + 0];
    float y = dg * verts[3 * i + 1] - nbr[3 * i + 1];
    float z = dg * verts[3 * i + 2] - nbr[3 * i + 2];
    nrm = sqrtf(x * x + y * y + z * z);
  }

  v2f A = { 1.0f, 1.0f };   // ones 16x4
  v2f B = { nrm, 0.0f };    // lane values in one B row per half-wave
  v8f C = {};
  C = __builtin_amdgcn_wmma_f32_16x16x4_f32(false, A, false, B,
                                            (short)0, C, false, false);
  float partial = C[0];     // lane L holds colsum_{L&15}
  partial += __shfl_xor(partial, 8, 32);
  partial += __shfl_xor(partial, 4, 32);
  partial += __shfl_xor(partial, 2, 32);
  partial += __shfl_xor(partial, 1, 32);

  __shared__ float smem[8]; // 256 threads = 8 waves on wave32
  int lane = threadIdx.x & 31;
  int wave = threadIdx.x >> 5;
  if (lane == 0) smem[wave] = partial;
  __syncthreads();
  if (threadIdx.x == 0) {
    float s = 0.0f;
    int nw = blockDim.x >> 5;
    for (int w = 0; w < nw; ++w) s += smem[w];
    atomicAdd(acc, s);
  }
}

__global__ void lap_writeout(const float* __restrict__ acc,
                             float* __restrict__ out, int Vn) {
  if (blockIdx.x == 0 && threadIdx.x == 0) out[0] = acc[0] / (float)Vn;
}

extern "C" void kernel_launch(void* const* d_in, const int* in_sizes, int n_in,
                              void* d_out, int out_size, void* d_ws, size_t ws_size,
                              hipStream_t stream) {
  const float* verts = (const float*)d_in[0];
  const int*   faces = (const int*)d_in[1];
  int Vn = in_sizes[0] / 3;   // 1,000,000
  int F  = in_sizes[1] / 3;   // 2,000,000

  char*  ws  = (char*)d_ws;
  float* nbr = (float*)ws;                 // 3V floats
  float* deg = nbr + 3ull * (size_t)Vn;    // V floats
  float* acc = deg + (size_t)Vn;           // 1 float
  size_t fcount = 4ull * (size_t)Vn + 8;   // floats to zero (incl. acc + pad)
  size_t hoff   = (fcount * sizeof(float) + 255) & ~(size_t)255;
  unsigned long long* htab = (unsigned long long*)(ws + hoff);

  // Size hash table to available workspace: prefer 2^26 slots, floor 2^24
  // (load factor 12M/16.7M = 0.72), never exceed ws_size.
  size_t avail = (ws_size > hoff) ? (ws_size - hoff) / 8 : 0;
  unsigned int H = 1u << 24;
  while ((size_t)(H << 1) <= avail && H < (1u << 26)) H <<= 1;
  while ((size_t)H > avail && H > (1u << 20)) H >>= 1;  // memory-safety clamp

  dim3 blk(256);
  lap_init<<<4096, blk, 0, stream>>>((float*)ws, (unsigned long long)fcount,
                                     htab, (unsigned long long)H);
  lap_scatter<<<(F + 255) / 256, blk, 0, stream>>>(verts, faces, F, Vn,
                                                   nbr, deg, htab, H - 1u);
  lap_finalize<<<(Vn + 255) / 256, blk, 0, stream>>>(verts, nbr, deg, acc, Vn);
  lap_writeout<<<1, 32, 0, stream>>>(acc, (float*)d_out, Vn);
}